// DetBenchEval_23055384445523
// MI455X (gfx1250) — compile-verified
//
#include <hip/hip_runtime.h>
#include <stdint.h>

#define NCLS      90
#define TOPK      5000
#define NDET      100
#define NBINS     16384
#define PSHIFT    18          // 32 - 14 bits
#define CAND_CAP  8192
#define NIMG      32

// ---------------- workspace layout (bytes) ----------------
#define HIST_OFF  0u                       // 32*16384*4 = 2,097,152
#define CNT_OFF   2097152u                 // 32*4
#define P_OFF     2097408u                 // 32*4
#define CAND_OFF  2097664u                 // 32*8192*8 = 2,097,152
#define TIDX_OFF  4194816u                 // 32*5000*4
#define TKEY_OFF  4834816u                 // 32*5000*4
#define TBOX_OFF  5474816u                 // 32*5000*16 (float4 aligned)
#define TSC_OFF   8034816u                 // 32*5000*4
#define TCLS_OFF  8674816u                 // 32*5000*4  (end ~9.3 MB)

__device__ __forceinline__ uint32_t fkey(float f) {
  uint32_t u = __float_as_uint(f);
  return u ^ (uint32_t)(((int32_t)u >> 31) | 0x80000000);
}

// ---------------- zero scratch ----------------
__global__ void k_zero(uint32_t* p, int n) {
  int i = blockIdx.x * blockDim.x + threadIdx.x;
  if (i < n) p[i] = 0u;
}

// ---------------- per-image histogram of key prefixes ----------------
__global__ void k_hist(const float4* __restrict__ cls, uint32_t* __restrict__ hist,
                       int imgQuads, int blocksPerImg) {
  __shared__ uint32_t h[NBINS];
  for (int i = threadIdx.x; i < NBINS; i += 256) h[i] = 0u;
  __syncthreads();
  int img = blockIdx.x / blocksPerImg;
  int blk = blockIdx.x % blocksPerImg;
  const float4* base = cls + (size_t)img * imgQuads;
  int q0 = blk * 4096;
  int q1 = min(q0 + 4096, imgQuads);
  for (int q = q0 + threadIdx.x; q < q1; q += 256) {
    __builtin_prefetch(base + q + 2048, 0, 0);   // global_prefetch_b8
    float4 v = base[q];
    atomicAdd(&h[fkey(v.x) >> PSHIFT], 1u);
    atomicAdd(&h[fkey(v.y) >> PSHIFT], 1u);
    atomicAdd(&h[fkey(v.z) >> PSHIFT], 1u);
    atomicAdd(&h[fkey(v.w) >> PSHIFT], 1u);
  }
  __syncthreads();
  uint32_t* gh = hist + (size_t)img * NBINS;
  for (int i = threadIdx.x; i < NBINS; i += 256)
    if (h[i]) atomicAdd(&gh[i], h[i]);
}

// ---------------- find threshold prefix per image ----------------
__global__ void k_thresh(const uint32_t* __restrict__ hist, uint32_t* __restrict__ P) {
  int img = blockIdx.x;
  if (threadIdx.x == 0) {
    const uint32_t* h = hist + (size_t)img * NBINS;
    uint32_t cum = 0;
    int bin = NBINS - 1;
    for (; bin > 0; --bin) { cum += h[bin]; if (cum >= TOPK) break; }
    P[img] = (uint32_t)bin;
  }
}

// ---------------- compact candidates (key, flat score index) ----------------
__global__ void k_compact(const float4* __restrict__ cls, int imgQuads, int blocksPerImg,
                          int logSq, int scoreBase, const uint32_t* __restrict__ P,
                          uint2* __restrict__ cand, uint32_t* __restrict__ cnt) {
  int img = blockIdx.x / blocksPerImg;
  int blk = blockIdx.x % blocksPerImg;
  uint32_t p = P[img];
  const float4* base = cls + (size_t)img * imgQuads;
  uint2* cbase = cand + (size_t)img * CAND_CAP;
  int q0 = blk * 4096;
  int q1 = min(q0 + 4096, imgQuads);
  int sqm1 = (1 << logSq) - 1;
  for (int q = q0 + threadIdx.x; q < q1; q += 256) {
    __builtin_prefetch(base + q + 2048, 0, 0);
    float4 v = base[q];
    float vv[4] = {v.x, v.y, v.z, v.w};
#pragma unroll
    for (int t = 0; t < 4; ++t) {
      uint32_t k = fkey(vv[t]);
      if ((k >> PSHIFT) >= p) {
        uint32_t slot = atomicAdd(&cnt[img], 1u);
        if (slot < CAND_CAP) {
          int m = q * 4 + t;                 // element index within image tensor
          uint32_t c = (uint32_t)m >> logSq; // channel
          uint32_t r = (uint32_t)m & sqm1;   // spatial h*s+w
          cbase[slot] = make_uint2(k, (uint32_t)scoreBase + r * (NCLS * 9u) + c);
        }
      }
    }
  }
}

// ---------------- exact top-k: bitonic sort of candidates in LDS ----------------
__global__ __launch_bounds__(512) void k_sort(const uint2* __restrict__ cand,
                                              const uint32_t* __restrict__ cnt,
                                              uint32_t* __restrict__ tidx,
                                              uint32_t* __restrict__ tkey) {
  __shared__ uint64_t shc[CAND_CAP];       // 64 KB
  int img = blockIdx.x;
  // Async bulk copy 64 KB of candidates global -> LDS (gfx1250 async path)
  {
    uint32_t ldsbase = (uint32_t)(uintptr_t)(&shc[0]);
    const char* g = (const char*)(cand + (size_t)img * CAND_CAP);
    for (int t = threadIdx.x; t < (CAND_CAP * 8) / 16; t += 512) {
      uint32_t la = ldsbase + (uint32_t)t * 16u;
      const void* ga = (const void*)(g + (size_t)t * 16);
      asm volatile("global_load_async_to_lds_b128 %0, %1, off"
                   :: "v"(la), "v"(ga) : "memory");
    }
    asm volatile("s_wait_asynccnt 0x0" ::: "memory");
  }
  __syncthreads();
  uint32_t n = cnt[img];
  if (n > CAND_CAP) n = CAND_CAP;
  // Build composite keys in place; pad stale slots deterministically.
  for (int i = threadIdx.x; i < CAND_CAP; i += 512) {
    uint64_t raw = shc[i];
    uint32_t k  = (uint32_t)raw;           // .x (low addr) = key
    uint32_t id = (uint32_t)(raw >> 32);   // .y = flat index
    uint64_t comp = ((uint64_t)(~k) << 32) | (uint64_t)id; // ascending => key desc, idx asc
    shc[i] = (i < (int)n) ? comp : 0xFFFFFFFFFFFFFFFFULL;
  }
  __syncthreads();
  // Bitonic sort ascending
  for (int kk = 2; kk <= CAND_CAP; kk <<= 1) {
    for (int j = kk >> 1; j > 0; j >>= 1) {
      for (int i = threadIdx.x; i < CAND_CAP; i += 512) {
        int ixj = i ^ j;
        if (ixj > i) {
          uint64_t a = shc[i], b = shc[ixj];
          bool up = ((i & kk) == 0);
          if ((a > b) == up) { shc[i] = b; shc[ixj] = a; }
        }
      }
      __syncthreads();
    }
  }
  for (int i = threadIdx.x; i < TOPK; i += 512) {
    uint64_t c = shc[i];
    tidx[(size_t)img * TOPK + i] = (uint32_t)c;
    tkey[(size_t)img * TOPK + i] = ~(uint32_t)(c >> 32);
  }
}

// ---------------- decode boxes + sigmoid scores ----------------
__global__ void k_decode(const float* __restrict__ b0, const float* __restrict__ b1,
                         const float* __restrict__ b2, const float* __restrict__ b3,
                         const float* __restrict__ b4,
                         const uint32_t* __restrict__ tidx, const uint32_t* __restrict__ tkey,
                         float4* __restrict__ tbox, float* __restrict__ tsc,
                         int* __restrict__ tcls) {
  int img = blockIdx.x / 20;
  int e = (blockIdx.x % 20) * 256 + threadIdx.x;
  if (e >= TOPK) return;
  size_t off = (size_t)img * TOPK + e;
  uint32_t sidx = tidx[off];
  uint32_t key  = tkey[off];
  uint32_t bits = (key & 0x80000000u) ? (key ^ 0x80000000u) : ~key;
  float logit = __uint_as_float(bits);
  float score = 1.0f / (1.0f + expf(-logit));
  int cls = (int)(sidx % NCLS);
  int anchor = (int)(sidx / NCLS);
  const int bases[6] = {0, 36864, 46080, 48384, 48960, 49104};
  int l = 0;
  while (anchor >= bases[l + 1]) ++l;
  int rel = anchor - bases[l];
  int a = rel % 9, r = rel / 9;
  int sdim = 64 >> l, sq = sdim * sdim;
  int hq = r / sdim, wq = r % sdim;
  int octave = a / 3, asp = a % 3;
  // anchor box in double, exactly like numpy, then single f32 rounding
  const double OCT[3]  = {1.0, 1.2599210498948731647672, 1.5874010519681994747517};
  const double ASPX[3] = {1.0, 1.4, 0.7};
  const double ASPY[3] = {1.0, 0.7, 1.4};
  double stride = (double)(8 << l);
  double base = (4.0 * stride) * OCT[octave];
  double ax2 = (base * ASPX[asp]) * 0.5;
  double ay2 = (base * ASPY[asp]) * 0.5;
  double xv = stride * 0.5 + (double)wq * stride;
  double yv = stride * 0.5 + (double)hq * stride;
  float Ay1 = (float)(yv - ay2);
  float Ax1 = (float)(xv - ax2);
  float Ay2 = (float)(yv + ay2);
  float Ax2 = (float)(xv + ax2);
  // gather regression values
  const float* bp;
  if (l == 0) bp = b0; else if (l == 1) bp = b1; else if (l == 2) bp = b2;
  else if (l == 3) bp = b3; else bp = b4;
  bp += (size_t)img * 36 * sq;
  float ty = bp[(a * 4 + 0) * sq + r];
  float tx = bp[(a * 4 + 1) * sq + r];
  float th = bp[(a * 4 + 2) * sq + r];
  float tw = bp[(a * 4 + 3) * sq + r];
  // decode in f32, no FMA contraction (match XLA op-by-op rounding)
  float yca = __fmul_rn(__fadd_rn(Ay1, Ay2), 0.5f);
  float xca = __fmul_rn(__fadd_rn(Ax1, Ax2), 0.5f);
  float ha  = __fsub_rn(Ay2, Ay1);
  float wa  = __fsub_rn(Ax2, Ax1);
  float w_  = __fmul_rn(expf(tw), wa);
  float h_  = __fmul_rn(expf(th), ha);
  float yc  = __fadd_rn(__fmul_rn(ty, ha), yca);
  float xc  = __fadd_rn(__fmul_rn(tx, wa), xca);
  float hw  = __fmul_rn(w_, 0.5f);
  float hh  = __fmul_rn(h_, 0.5f);
  float4 bb;
  bb.x = __fsub_rn(xc, hw);
  bb.y = __fsub_rn(yc, hh);
  bb.z = __fadd_rn(xc, hw);
  bb.w = __fadd_rn(yc, hh);
  tbox[off] = bb;
  tsc[off] = score;
  tcls[off] = cls;
}

// ---------------- greedy NMS (one workgroup per image) ----------------
__global__ __launch_bounds__(1024) void k_nms(const float4* __restrict__ tbox,
                                              const float* __restrict__ tsc,
                                              const int* __restrict__ tcls,
                                              const float* __restrict__ scales,
                                              float* __restrict__ out) {
  __shared__ float sx1[TOPK], sy1[TOPK], sx2[TOPK], sy2[TOPK], sar[TOPK], ssc[TOPK];
  __shared__ float rv[1024];
  __shared__ int   ri[1024];
  __shared__ int   keepI[NDET];
  __shared__ float keptV[NDET];
  __shared__ int   selIdx;
  int img = blockIdx.x;
  int tid = threadIdx.x;
  // load + class offset + areas (replicating reference arithmetic exactly)
  for (int j = tid; j < TOPK; j += 1024) {
    size_t off = (size_t)img * TOPK + j;
    float4 bb = tbox[off];
    float co = __fmul_rn(__int2float_rn(tcls[off]), 8192.0f); // 512*16
    float x1 = __fadd_rn(bb.x, co);
    float y1 = __fadd_rn(bb.y, co);
    float x2 = __fadd_rn(bb.z, co);
    float y2 = __fadd_rn(bb.w, co);
    sx1[j] = x1; sy1[j] = y1; sx2[j] = x2; sy2[j] = y2;
    sar[j] = __fmul_rn(__fsub_rn(x2, x1), __fsub_rn(y2, y1));
    ssc[j] = tsc[off];
  }
  __syncthreads();
  for (int it = 0; it < NDET; ++it) {
    // argmax with first-occurrence tie-break (matches jnp.argmax over same ordering)
    float bv = -3.402823466e+38f;
    int bi = 0x7FFFFFFF;
    for (int j = tid; j < TOPK; j += 1024) {
      float v = ssc[j];
      if (v > bv || (v == bv && j < bi)) { bv = v; bi = j; }
    }
    rv[tid] = bv; ri[tid] = bi;
    __syncthreads();
    for (int s = 512; s > 0; s >>= 1) {
      if (tid < s) {
        float v2 = rv[tid + s]; int i2 = ri[tid + s];
        if (v2 > rv[tid] || (v2 == rv[tid] && i2 < ri[tid])) { rv[tid] = v2; ri[tid] = i2; }
      }
      __syncthreads();
    }
    if (tid == 0) {
      selIdx = ri[0];
      keepI[it] = ri[0];
      keptV[it] = rv[0];
    }
    __syncthreads();
    int sel = selIdx;
    float X1 = sx1[sel], Y1 = sy1[sel], X2 = sx2[sel], Y2 = sy2[sel], AR = sar[sel];
    for (int j = tid; j < TOPK; j += 1024) {
      float xx1 = fmaxf(sx1[j], X1);
      float yy1 = fmaxf(sy1[j], Y1);
      float xx2 = fminf(sx2[j], X2);
      float yy2 = fminf(sy2[j], Y2);
      float iw = fmaxf(__fsub_rn(xx2, xx1), 0.0f);
      float ih = fmaxf(__fsub_rn(yy2, yy1), 0.0f);
      float inter = __fmul_rn(iw, ih);
      float denom = __fsub_rn(__fadd_rn(sar[j], AR), inter);
      float iou = inter / denom;
      if (iou > 0.5f || j == sel) ssc[j] = -1.0f;
    }
    __syncthreads();
  }
  // emit detections
  if (tid < NDET) {
    float kv = keptV[tid];
    bool valid = kv > 0.0f;
    int k = keepI[tid];
    float sc = scales[img];
    float4 bb = tbox[(size_t)img * TOPK + k];
    int c = tcls[(size_t)img * TOPK + k];
    float* row = out + ((size_t)img * NDET + tid) * 6;
    row[0] = valid ? __fmul_rn(bb.x, sc) : 0.0f;
    row[1] = valid ? __fmul_rn(bb.y, sc) : 0.0f;
    row[2] = valid ? __fmul_rn(bb.z, sc) : 0.0f;
    row[3] = valid ? __fmul_rn(bb.w, sc) : 0.0f;
    row[4] = valid ? kv : 0.0f;
    row[5] = valid ? (float)c : 0.0f;
  }
}

extern "C" void kernel_launch(void* const* d_in, const int* in_sizes, int n_in,
                              void* d_out, int out_size, void* d_ws, size_t ws_size,
                              hipStream_t stream) {
  // setup_inputs() dict order: cls_l3, box_l3, cls_l4, box_l4, ..., cls_l7, box_l7, image_scales
  const float* cls[5];
  const float* box[5];
  for (int l = 0; l < 5; ++l) {
    cls[l] = (const float*)d_in[2 * l];
    box[l] = (const float*)d_in[2 * l + 1];
  }
  const float* scales = (const float*)d_in[10];
  char* ws = (char*)d_ws;
  uint32_t* hist = (uint32_t*)(ws + HIST_OFF);
  uint32_t* cnt  = (uint32_t*)(ws + CNT_OFF);
  uint32_t* P    = (uint32_t*)(ws + P_OFF);
  uint2*    cand = (uint2*)(ws + CAND_OFF);
  uint32_t* tidx = (uint32_t*)(ws + TIDX_OFF);
  uint32_t* tkey = (uint32_t*)(ws + TKEY_OFF);
  float4*   tbox = (float4*)(ws + TBOX_OFF);
  float*    tsc  = (float*)(ws + TSC_OFF);
  int*      tcls = (int*)(ws + TCLS_OFF);

  // zero histograms + counters (everything up to CAND_OFF)
  {
    int nwords = (int)(CAND_OFF / 4);
    k_zero<<<(nwords + 255) / 256, 256, 0, stream>>>((uint32_t*)ws, nwords);
  }

  const int sdims[5] = {64, 32, 16, 8, 4};
  const int scoreBase[5] = {0, 3317760, 4147200, 4354560, 4406400};
  int imgQuads[5], bpi[5], logSq[5];
  for (int l = 0; l < 5; ++l) {
    int sq = sdims[l] * sdims[l];
    imgQuads[l] = (NCLS * 9 * sq) / 4;               // 810*s*s / 4
    bpi[l] = (imgQuads[l] + 4095) / 4096;
    int ls = 0; while ((1 << ls) < sq) ++ls;
    logSq[l] = ls;
  }
  // histogram passes
  for (int l = 0; l < 5; ++l)
    k_hist<<<NIMG * bpi[l], 256, 0, stream>>>((const float4*)cls[l], hist, imgQuads[l], bpi[l]);
  // threshold prefix per image
  k_thresh<<<NIMG, 32, 0, stream>>>(hist, P);
  // compaction passes
  for (int l = 0; l < 5; ++l)
    k_compact<<<NIMG * bpi[l], 256, 0, stream>>>((const float4*)cls[l], imgQuads[l], bpi[l],
                                                 logSq[l], scoreBase[l], P, cand, cnt);
  // exact top-k via bitonic sort (async LDS fill)
  k_sort<<<NIMG, 512, 0, stream>>>(cand, cnt, tidx, tkey);
  // decode boxes + scores
  k_decode<<<NIMG * 20, 256, 0, stream>>>(box[0], box[1], box[2], box[3], box[4],
                                          tidx, tkey, tbox, tsc, tcls);
  // per-image greedy NMS + output
  k_nms<<<NIMG, 1024, 0, stream>>>(tbox, tsc, tcls, scales, (float*)d_out);
}